// EntityPredictor_89163521065360
// MI455X (gfx1250) — compile-verified
//
#include <hip/hip_runtime.h>
#include <hip/hip_bf16.h>
#include <math.h>

#define BSZ    64
#define SEQ    512
#define DD     768
#define NSPANS 1280
#define LL     5
#define HH     200
#define NENT   16
#define GATES  (4*HH)        // 800
#define NCOLS  (2*GATES)     // 1600 (fwd gates | bwd gates)
#define MROWS  (NSPANS*LL)   // 6400

#define KC     32            // K-chunk staged in LDS per iteration
#define PITCH  36            // LDS row pitch (floats); 36 -> conflict-free ds_load_b64
#define NCHUNK (DD/KC)       // 24

typedef float v2f __attribute__((ext_vector_type(2)));
typedef float v8f __attribute__((ext_vector_type(8)));

__device__ __forceinline__ v8f wmma4(v2f a, v2f b, v8f c) {
  // V_WMMA_F32_16X16X4_F32 : D(16x16,f32) = A(16x4,f32) * B(4x16,f32) + C
  return __builtin_amdgcn_wmma_f32_16x16x4_f32(false, a, false, b, (short)0, c,
                                               false, false);
}

// Async global->LDS copy (gfx1250, ASYNCcnt-tracked). vdst = LDS byte offset.
__device__ __forceinline__ void async_copy_b128(const float* g, unsigned lds_off) {
  asm volatile("global_load_async_to_lds_b128 %0, %1, off"
               :: "v"(lds_off), "v"(g) : "memory");
}
__device__ __forceinline__ void wait_async0() {
  asm volatile("s_wait_asynccnt 0x0" ::: "memory");
}

// ---------------------------------------------------------------------------
// Kernel 1: XG = gather(hidden)[6400 x 768] @ [W_ih_f | W_ih_b]^T + (b_ih+b_hh)
// 4-wave workgroup computes a 256x64 tile; the shared 64-col B panel is staged
// through LDS with double-buffered async global->LDS copies (shared by all 4
// waves), A gather is fused into per-wave global addressing.
// ---------------------------------------------------------------------------
__global__ void __launch_bounds__(128)
input_gemm(const float* __restrict__ hidden,
           const int*   __restrict__ tok_idx,     // (NSPANS, LL)
           const int*   __restrict__ span_batch,  // (NSPANS)
           const float* __restrict__ Wf,          // (800, 768)
           const float* __restrict__ Wb,          // (800, 768)
           const float* __restrict__ bihf, const float* __restrict__ bhhf,
           const float* __restrict__ bihb, const float* __restrict__ bhhb,
           float* __restrict__ XG)                // (6400, 1600)
{
  __shared__ float Bsm[2][64 * PITCH];            // 2 x 2304 floats = 18 KB

  const int tid  = threadIdx.x;
  const int wave = tid >> 5;
  const int l    = tid & 31;
  const int lm   = l & 15;
  const int hf   = l >> 4;                        // half-wave K-pair select
  const int m0   = blockIdx.y * 256 + wave * 64;
  const int n0   = blockIdx.x * 64;

  // ---- staging setup: thread (col = tid/2, half = tid&1) covers 16 ks ----
  const int scol = tid >> 1;
  const int shalf = tid & 1;
  const int colg = n0 + scol;
  const float* wstage = (colg < GATES) ? (Wf + (size_t)colg * DD)
                                       : (Wb + (size_t)(colg - GATES) * DD);
  const unsigned ldsBase0 = (unsigned)(size_t)&Bsm[0][0];
  const unsigned ldsBase1 = (unsigned)(size_t)&Bsm[1][0];
  const unsigned sbase = (unsigned)((scol * PITCH + shalf * 16) * 4);

  // ---- per-wave A (gathered rows) and bias setup ----
  size_t abase[4];
  float bias[4];
#pragma unroll
  for (int mt = 0; mt < 4; ++mt) {
    int row = m0 + mt * 16 + lm;
    int nsp = row / LL;
    int t   = row % LL;
    int bb  = span_batch[nsp];
    int tok = tok_idx[nsp * LL + t];
    abase[mt] = ((size_t)bb * SEQ + (size_t)tok) * DD;
  }
#pragma unroll
  for (int nt = 0; nt < 4; ++nt) {
    int col = n0 + nt * 16 + lm;
    bias[nt] = (col < GATES) ? (bihf[col] + bhhf[col])
                             : (bihb[col - GATES] + bhhb[col - GATES]);
  }

  v8f acc[16];
  const v8f vzero = {0.f, 0.f, 0.f, 0.f, 0.f, 0.f, 0.f, 0.f};
#pragma unroll
  for (int i = 0; i < 16; ++i) acc[i] = vzero;

  // ---- prologue: stage chunk 0 into buffer 0 ----
  {
    const float* g = wstage + shalf * 16;
#pragma unroll
    for (int i = 0; i < 4; ++i)
      async_copy_b128(g + i * 4, ldsBase0 + sbase + i * 16);
  }
  wait_async0();
  __syncthreads();

  for (int c = 0; c < NCHUNK; ++c) {
    const int cur = c & 1;
    // stage next chunk into the other buffer (overlaps with WMMA below)
    if (c + 1 < NCHUNK) {
      const unsigned dstBase = (cur ? ldsBase0 : ldsBase1) + sbase;
      const float* g = wstage + (c + 1) * KC + shalf * 16;
#pragma unroll
      for (int i = 0; i < 4; ++i)
        async_copy_b128(g + i * 4, dstBase + i * 16);
    }

    const float* bb = &Bsm[cur][0];
#pragma unroll
    for (int kl = 0; kl < KC; kl += 4) {
      const int k = c * KC + kl;
      v2f a[4], b[4];
#pragma unroll
      for (int mt = 0; mt < 4; ++mt)
        a[mt] = *(const v2f*)(hidden + abase[mt] + k + 2 * hf);
#pragma unroll
      for (int nt = 0; nt < 4; ++nt)
        b[nt] = *(const v2f*)(bb + (nt * 16 + lm) * PITCH + kl + 2 * hf);
#pragma unroll
      for (int mt = 0; mt < 4; ++mt)
#pragma unroll
        for (int nt = 0; nt < 4; ++nt)
          acc[mt * 4 + nt] = wmma4(a[mt], b[nt], acc[mt * 4 + nt]);
    }

    wait_async0();       // own async writes to next buffer done
    __syncthreads();     // all waves done reading cur + staging next
  }

#pragma unroll
  for (int mt = 0; mt < 4; ++mt) {
#pragma unroll
    for (int nt = 0; nt < 4; ++nt) {
      v8f cacc = acc[mt * 4 + nt];
      const int ncol = n0 + nt * 16 + lm;
      const float bs = bias[nt];
#pragma unroll
      for (int r = 0; r < 8; ++r) {
        int mrow = m0 + mt * 16 + r + 8 * hf;
        XG[(size_t)mrow * NCOLS + ncol] = cacc[r] + bs;
      }
    }
  }
}

// ---------------------------------------------------------------------------
// Kernel 2: G(1280x800) = Hstate(1280x200) @ W_hh^T(200x800). One wave per
// 64x64 tile; N=800 edge tiles clamped (loads) / guarded (stores).
// ---------------------------------------------------------------------------
__global__ void __launch_bounds__(32)
rec_gemm(const float* __restrict__ Hst,   // (1280, 200)
         const float* __restrict__ Whh,   // (800, 200)
         float* __restrict__ G)           // (1280, 800)
{
  const int l  = threadIdx.x;
  const int lm = l & 15;
  const int hf = l >> 4;
  const int m0 = blockIdx.y * 64;
  const int n0 = blockIdx.x * 64;

  const float* aptr[4];
  const float* bptr[4];
#pragma unroll
  for (int mt = 0; mt < 4; ++mt)
    aptr[mt] = Hst + (size_t)(m0 + mt * 16 + lm) * HH;
#pragma unroll
  for (int nt = 0; nt < 4; ++nt) {
    int col = n0 + nt * 16 + lm;
    if (col >= GATES) col = GATES - 1;       // clamp (store is guarded)
    bptr[nt] = Whh + (size_t)col * HH;
  }

  v8f acc[16];
  const v8f vzero = {0.f, 0.f, 0.f, 0.f, 0.f, 0.f, 0.f, 0.f};
#pragma unroll
  for (int i = 0; i < 16; ++i) acc[i] = vzero;

  for (int k = 0; k < HH; k += 4) {
    v2f a[4], b[4];
#pragma unroll
    for (int mt = 0; mt < 4; ++mt)
      a[mt] = *(const v2f*)(aptr[mt] + k + 2 * hf);
#pragma unroll
    for (int nt = 0; nt < 4; ++nt)
      b[nt] = *(const v2f*)(bptr[nt] + k + 2 * hf);
#pragma unroll
    for (int mt = 0; mt < 4; ++mt)
#pragma unroll
      for (int nt = 0; nt < 4; ++nt)
        acc[mt * 4 + nt] = wmma4(a[mt], b[nt], acc[mt * 4 + nt]);
  }

#pragma unroll
  for (int mt = 0; mt < 4; ++mt) {
#pragma unroll
    for (int nt = 0; nt < 4; ++nt) {
      const int ncol = n0 + nt * 16 + lm;
      if (ncol < GATES) {
        v8f c = acc[mt * 4 + nt];
#pragma unroll
        for (int r = 0; r < 8; ++r) {
          int mrow = m0 + mt * 16 + r + 8 * hf;
          G[(size_t)mrow * GATES + ncol] = c[r];
        }
      }
    }
  }
}

// ---------------------------------------------------------------------------
// Kernel 3: fused LSTM gate update for step t (both directions).
// ---------------------------------------------------------------------------
__global__ void gate_step(const float* __restrict__ XG,
                          const float* __restrict__ Gf, const float* __restrict__ Gb,
                          float* __restrict__ Hf, float* __restrict__ Hb,
                          float* __restrict__ Cf, float* __restrict__ Cb,
                          float* __restrict__ Sf, float* __restrict__ Sb,
                          const int* __restrict__ span_len, int t)
{
  const int total = 2 * NSPANS * HH;
  int gid = blockIdx.x * blockDim.x + threadIdx.x;
  if (gid >= total) return;
  const int dir = (gid >= NSPANS * HH) ? 1 : 0;
  int r = gid - dir * NSPANS * HH;
  const int n = r / HH;
  const int j = r - n * HH;
  const int len = span_len[n];

  const float* G;
  float *Hp, *Cp, *Sp;
  int xgrow, coff;
  if (dir == 0) {
    G = Gf + (size_t)n * GATES; Hp = Hf; Cp = Cf; Sp = Sf;
    xgrow = n * LL + t; coff = 0;
  } else {
    int tb = len - 1 - t; if (tb < 0) tb = 0;
    G = Gb + (size_t)n * GATES; Hp = Hb; Cp = Cb; Sp = Sb;
    xgrow = n * LL + tb; coff = GATES;
  }
  const float* xg = XG + (size_t)xgrow * NCOLS + coff;

  float gi = G[j]          + xg[j];
  float gf = G[HH + j]     + xg[HH + j];
  float gg = G[2 * HH + j] + xg[2 * HH + j];
  float go = G[3 * HH + j] + xg[3 * HH + j];

  float si = 1.f / (1.f + expf(-gi));
  float sf = 1.f / (1.f + expf(-gf));
  float so = 1.f / (1.f + expf(-go));

  const size_t idx = (size_t)n * HH + j;
  float c  = Cp[idx];
  float cn = sf * c + si * tanhf(gg);
  float hn = so * tanhf(cn);
  Cp[idx] = cn;
  Hp[idx] = hn;
  if (t < len) Sp[idx] += hn;
}

// ---------------------------------------------------------------------------
// Kernel 4: logits = [sum_f | sum_b](1280x400) @ entity_embs^T(400x16)
// ---------------------------------------------------------------------------
__global__ void logits_kernel(const float* __restrict__ Sf,
                              const float* __restrict__ Sb,
                              const float* __restrict__ E,   // (16, 400)
                              float* __restrict__ out)       // (1280, 16)
{
  int gid = blockIdx.x * blockDim.x + threadIdx.x;
  if (gid >= NSPANS * NENT) return;
  const int n = gid / NENT;
  const int e = gid - n * NENT;
  const float* er = E + (size_t)e * (2 * HH);
  const float* sf = Sf + (size_t)n * HH;
  const float* sb = Sb + (size_t)n * HH;
  float acc = 0.f;
  for (int j = 0; j < HH; ++j) acc += sf[j] * er[j];
  for (int j = 0; j < HH; ++j) acc += sb[j] * er[HH + j];
  out[gid] = acc;
}

__global__ void zero_kernel(float* __restrict__ p, int count) {
  int gid = blockIdx.x * blockDim.x + threadIdx.x;
  if (gid < count) p[gid] = 0.f;
}

// ---------------------------------------------------------------------------
extern "C" void kernel_launch(void* const* d_in, const int* in_sizes, int n_in,
                              void* d_out, int out_size, void* d_ws, size_t ws_size,
                              hipStream_t stream)
{
  const float* hidden     = (const float*)d_in[0];
  const int*   tok_idx    = (const int*)  d_in[1];
  const int*   span_len   = (const int*)  d_in[2];
  const int*   span_batch = (const int*)  d_in[3];
  const float* Wihf       = (const float*)d_in[4];
  const float* Whhf       = (const float*)d_in[5];
  const float* bihf       = (const float*)d_in[6];
  const float* bhhf       = (const float*)d_in[7];
  const float* Wihb       = (const float*)d_in[8];
  const float* Whhb       = (const float*)d_in[9];
  const float* bihb       = (const float*)d_in[10];
  const float* bhhb       = (const float*)d_in[11];
  const float* E          = (const float*)d_in[12];
  float* out = (float*)d_out;

  // Workspace layout (floats): ~55.3 MB total
  float* ws = (float*)d_ws;
  float* XG = ws;                                  // 6400*1600
  float* Gf = XG + (size_t)MROWS * NCOLS;          // 1280*800
  float* Gb = Gf + (size_t)NSPANS * GATES;         // 1280*800
  float* Hf = Gb + (size_t)NSPANS * GATES;         // 6 x 1280*200
  float* Hb = Hf + (size_t)NSPANS * HH;
  float* Cf = Hb + (size_t)NSPANS * HH;
  float* Cb = Cf + (size_t)NSPANS * HH;
  float* Sf = Cb + (size_t)NSPANS * HH;
  float* Sb = Sf + (size_t)NSPANS * HH;

  const int zcount = 6 * NSPANS * HH;              // H, C, S for both dirs
  zero_kernel<<<(zcount + 255) / 256, 256, 0, stream>>>(Hf, zcount);

  dim3 g1(NCOLS / 64, MROWS / 256);                // 25 x 25 WGs, 4 waves each
  input_gemm<<<g1, 128, 0, stream>>>(hidden, tok_idx, span_batch,
                                     Wihf, Wihb, bihf, bhhf, bihb, bhhb, XG);

  dim3 g2((GATES + 63) / 64, NSPANS / 64);         // 13 x 20 waves
  const int gthreads = 2 * NSPANS * HH;
  for (int t = 0; t < LL; ++t) {
    rec_gemm<<<g2, 32, 0, stream>>>(Hf, Whhf, Gf);
    rec_gemm<<<g2, 32, 0, stream>>>(Hb, Whhb, Gb);
    gate_step<<<(gthreads + 255) / 256, 256, 0, stream>>>(
        XG, Gf, Gb, Hf, Hb, Cf, Cb, Sf, Sb, span_len, t);
  }

  logits_kernel<<<(NSPANS * NENT + 255) / 256, 256, 0, stream>>>(Sf, Sb, E, out);
}